// GritTransformerLayer_19464791785729
// MI455X (gfx1250) — compile-verified
//
// GRIT transformer layer for MI455X (gfx1250), fp32 throughout.
// Dense GEMMs: V_WMMA_F32_16X16X4_F32 (exact fp32, matches reference numerics).
//   - one wave per 16-row strip: A loaded to registers ONCE (b64 clause), reused
//     across all column tiles (8x HBM traffic cut on the E x 128 edge GEMM)
//   - W staged transposed in LDS (pad K+2): B fragments via ds_load_2addr_b64
//   - wave-uniform full-strip fast path: unguarded, coalesced store clauses
// Workspace requirement: ~951 MB (Ee E*128 floats dominates).
#include <hip/hip_runtime.h>
#include <math.h>

#define DD 64   // feature dim
#define HH 8    // heads
#define DH 8    // head dim
#define CLAMPV 5.0f

typedef float v2f __attribute__((ext_vector_type(2)));
typedef float v8f __attribute__((ext_vector_type(8)));

// ---------------------------------------------------------------- utilities
__global__ void fill_kernel(float* __restrict__ p, float v, int n) {
    int i = blockIdx.x * blockDim.x + threadIdx.x;
    if (i < n) p[i] = v;
}

__global__ void deg_kernel(const int* __restrict__ ei, float* __restrict__ deg, int E) {
    int e = blockIdx.x * blockDim.x + threadIdx.x;
    if (e < E) atomicAdd(&deg[ei[E + e]], 1.0f);
}

__device__ __forceinline__ void atomicMaxFloat(float* addr, float val) {
    unsigned int* ua = (unsigned int*)addr;
    unsigned int old = __float_as_uint(*addr);
    while (__uint_as_float(old) < val) {
        unsigned int assumed = old;
        old = atomicCAS(ua, assumed, __float_as_uint(val));
        if (old == assumed) break;
    }
}

// ---------------------------------------------------------------- WMMA GEMM
// C[M x NC] = act(A[M x K] @ W[K x NC] + bias).
// One wave per 16-row strip of C; 8 waves (256 threads) per block.
// fp32 WMMA 16x16x4 lane layout:
//   A: row = lane&15, K-pair {kb,kb+1}, kb = 2*(lane>>4)
//   B: col = lane&15, same K-pair
//   C/D: lane l, vgpr v -> row = v + 8*(l>>4), col = l&15
template<int K, int NC, bool BIAS, bool RELU>
__global__ void __launch_bounds__(256) gemm_wmma(
        const float* __restrict__ A, const float* __restrict__ W,
        const float* __restrict__ bias, float* __restrict__ C, int M) {
    constexpr int KP = K + 2;                    // pad: 8B-aligned k-pairs, no bank conflicts
    __shared__ float sW[NC * KP];

    // ---- cooperative staging: sW[col*KP + k] = W[k*NC + col] (transposed) ----
    for (int i = threadIdx.x; i < K * NC; i += 256) {
        int k = i / NC, c = i % NC;              // consecutive threads: consecutive cols
        sW[c * KP + k] = W[i];
    }
    __syncthreads();

    const int lane = threadIdx.x & 31;
    const int wave = (blockIdx.x * blockDim.x + threadIdx.x) >> 5;
    const int mtiles = (M + 15) >> 4;
    if (wave >= mtiles) return;                  // wave-uniform: EXEC stays all-1 for WMMA

    const bool full = (wave * 16 + 16) <= M;     // wave-uniform: strip fully in-range
    int arow = wave * 16 + (lane & 15);
    if (arow >= M) arow = M - 1;                 // clamp tail rows (tail stores guarded)
    const int kb = (lane >> 4) << 1;

    // ---- load A strip once into registers, reuse for every column tile ----
    v2f a[K / 4];
#pragma unroll
    for (int s = 0; s < K / 4; ++s) {
        a[s].x = A[(size_t)arow * K + s * 4 + kb];
        a[s].y = A[(size_t)arow * K + s * 4 + kb + 1];
    }

    const int rbase = wave * 16 + ((lane >> 4) << 3);
#pragma unroll
    for (int nt = 0; nt < NC / 16; ++nt) {
        const int col = nt * 16 + (lane & 15);
        v8f acc = {};
#pragma unroll
        for (int s = 0; s < K / 4; ++s) {
            // (col*KP + s*4 + kb)*4 bytes is 8-byte aligned: KP even, k even
            v2f b = *(const v2f*)&sW[col * KP + s * 4 + kb];   // ds_load_b64
            acc = __builtin_amdgcn_wmma_f32_16x16x4_f32(false, a[s], false, b,
                                                        (short)0, acc, false, false);
        }
        const float bv = BIAS ? bias[col] : 0.0f;
        if (full) {                               // fast path: straight-line store clause
#pragma unroll
            for (int v = 0; v < 8; ++v) {
                float val = acc[v] + bv;
                if (RELU) val = fmaxf(val, 0.0f);
                C[(size_t)(rbase + v) * NC + col] = val;
            }
        } else {                                  // generic tail strip (unused when M%16==0)
#pragma unroll
            for (int v = 0; v < 8; ++v) {
                int r = rbase + v;
                if (r < M) {
                    float val = acc[v] + bv;
                    if (RELU) val = fmaxf(val, 0.0f);
                    C[(size_t)r * NC + col] = val;
                }
            }
        }
    }
}

// ---------------------------------------------------------------- edge stage
// s = relu( signed_sqrt((K[src]+Q[dst]) * Ew) + Eb );  wE = s  (E x 64)
__global__ void edge_s_kernel(const int* __restrict__ ei, const float* __restrict__ Q,
                              const float* __restrict__ Kf, const float* __restrict__ Ee,
                              float* __restrict__ wE, int E) {
    int t = blockIdx.x * blockDim.x + threadIdx.x;
    if (t >= E * DD) return;
    int e = t >> 6, idx = t & 63;
    int h = idx >> 3, d = idx & 7;
    int src = ei[e], dst = ei[E + e];
    float kq = Kf[src * DD + idx] + Q[dst * DD + idx];
    float ew = Ee[e * 128 + h * 16 + d];
    float eb = Ee[e * 128 + h * 16 + 8 + d];
    float s = kq * ew;
    s = sqrtf(fmaxf(s, 0.f)) - sqrtf(fmaxf(-s, 0.f));  // signed sqrt
    wE[t] = fmaxf(s + eb, 0.f);
}

__device__ __forceinline__ float edge_score(const float* __restrict__ wE,
                                            const float* __restrict__ Aw, int e, int h) {
    float sc = 0.f;
#pragma unroll
    for (int d = 0; d < DH; ++d)
        sc += wE[e * DD + h * DH + d] * Aw[d * HH + h];   // Aw (Dh,H,1)
    return fminf(fmaxf(sc, -CLAMPV), CLAMPV);
}

__global__ void score_max_kernel(const int* __restrict__ ei, const float* __restrict__ wE,
                                 const float* __restrict__ Aw, float* __restrict__ m, int E) {
    int t = blockIdx.x * blockDim.x + threadIdx.x;
    if (t >= E * HH) return;
    int e = t >> 3, h = t & 7;
    atomicMaxFloat(&m[ei[E + e] * HH + h], edge_score(wE, Aw, e, h));
}

__global__ void m_fix_kernel(float* __restrict__ m, int n) {
    int t = blockIdx.x * blockDim.x + threadIdx.x;
    if (t < n && !isfinite(m[t])) m[t] = 0.f;
}

__global__ void denom_kernel(const int* __restrict__ ei, const float* __restrict__ wE,
                             const float* __restrict__ Aw, const float* __restrict__ m,
                             float* __restrict__ denom, int E) {
    int t = blockIdx.x * blockDim.x + threadIdx.x;
    if (t >= E * HH) return;
    int e = t >> 3, h = t & 7;
    int dst = ei[E + e];
    float sc = edge_score(wE, Aw, e, h);
    atomicAdd(&denom[dst * HH + h], expf(sc - m[dst * HH + h]));
}

__global__ void aggregate_kernel(const int* __restrict__ ei, const float* __restrict__ wE,
                                 const float* __restrict__ Aw, const float* __restrict__ m,
                                 const float* __restrict__ denom, const float* __restrict__ V,
                                 float* __restrict__ wV, float* __restrict__ rowV, int E) {
    int t = blockIdx.x * blockDim.x + threadIdx.x;
    if (t >= E * HH) return;
    int e = t >> 3, h = t & 7;
    int src = ei[e], dst = ei[E + e];
    float sc = edge_score(wE, Aw, e, h);
    float attn = expf(sc - m[dst * HH + h]) / (denom[dst * HH + h] + 1e-16f);
#pragma unroll
    for (int d = 0; d < DH; ++d) {
        int idx = h * DH + d;
        atomicAdd(&wV[dst * DD + idx], V[src * DD + idx] * attn);
        atomicAdd(&rowV[dst * DD + idx], wE[e * DD + idx] * attn);
    }
}

// wV += einsum('nhd,dhc->nhc', rowV, VeRow); then degree scaling -> hmid
__global__ void node_kernel(const float* __restrict__ wV, const float* __restrict__ rowV,
                            const float* __restrict__ VeRow, const float* __restrict__ deg,
                            const float* __restrict__ deg_coef, float* __restrict__ hmid, int N) {
    int t = blockIdx.x * blockDim.x + threadIdx.x;
    if (t >= N * DD) return;
    int n = t >> 6, idx = t & 63, h = idx >> 3, c = idx & 7;
    float acc = wV[t];
#pragma unroll
    for (int d = 0; d < DH; ++d)
        acc += rowV[n * DD + h * DH + d] * VeRow[d * DD + h * DH + c];  // VeRow (Dh,H,Dh)
    float ld = logf(deg[n] + 1.0f);
    hmid[t] = acc * deg_coef[idx * 2] + acc * ld * deg_coef[idx * 2 + 1];
}

// ---------------------------------------------------------------- batch-norm
// v = a + b (residual); stats[0:64]=sum, [64:128]=sumsq, [128:192]=mu, [192:256]=rstd
__global__ void bn_stats_kernel(const float* __restrict__ a, const float* __restrict__ b,
                                float* __restrict__ stats, int total) {
    __shared__ float ss[DD], sq[DD];
    if (threadIdx.x < DD) { ss[threadIdx.x] = 0.f; sq[threadIdx.x] = 0.f; }
    __syncthreads();
    float s = 0.f, q = 0.f;
    int c = threadIdx.x & 63;  // stride is a multiple of 64 -> column fixed per thread
    for (int i = blockIdx.x * blockDim.x + threadIdx.x; i < total; i += gridDim.x * blockDim.x) {
        float v = a[i] + b[i];
        s += v; q += v * v;
    }
    atomicAdd(&ss[c], s); atomicAdd(&sq[c], q);
    __syncthreads();
    if (threadIdx.x < DD) {
        atomicAdd(&stats[threadIdx.x], ss[threadIdx.x]);
        atomicAdd(&stats[DD + threadIdx.x], sq[threadIdx.x]);
    }
}

__global__ void bn_finalize_kernel(float* __restrict__ stats, float invM) {
    int c = threadIdx.x;
    if (c < DD) {
        float mu  = stats[c] * invM;
        float var = stats[DD + c] * invM - mu * mu;
        stats[2 * DD + c] = mu;
        stats[3 * DD + c] = rsqrtf(fmaxf(var, 0.f) + 1e-5f);
    }
}

__global__ void bn_apply_kernel(const float* __restrict__ a, const float* __restrict__ b,
                                const float* __restrict__ stats, const float* __restrict__ gamma,
                                const float* __restrict__ beta, float* __restrict__ out, int total) {
    int i = blockIdx.x * blockDim.x + threadIdx.x;
    if (i >= total) return;
    int c = i & 63;
    float v = a[i] + b[i];
    out[i] = (v - stats[2 * DD + c]) * stats[3 * DD + c] * gamma[c] + beta[c];
}

// ---------------------------------------------------------------- driver
static inline int eb(int n) { return (n + 255) / 256; }

extern "C" void kernel_launch(void* const* d_in, const int* in_sizes, int n_in,
                              void* d_out, int out_size, void* d_ws, size_t ws_size,
                              hipStream_t stream) {
    const float* x         = (const float*)d_in[0];
    const float* edge_attr = (const float*)d_in[1];
    const int*   ei        = (const int*)  d_in[2];
    const float* Wq  = (const float*)d_in[3];  const float* bq  = (const float*)d_in[4];
    const float* Wk  = (const float*)d_in[5];
    const float* We  = (const float*)d_in[6];  const float* be  = (const float*)d_in[7];
    const float* Wv  = (const float*)d_in[8];
    const float* Aw  = (const float*)d_in[9];
    const float* VeRow    = (const float*)d_in[10];
    const float* deg_coef = (const float*)d_in[11];
    const float* WOh = (const float*)d_in[12]; const float* bOh = (const float*)d_in[13];
    const float* WOe = (const float*)d_in[14]; const float* bOe = (const float*)d_in[15];
    const float* g1h = (const float*)d_in[16]; const float* be1h = (const float*)d_in[17];
    const float* g1e = (const float*)d_in[18]; const float* be1e = (const float*)d_in[19];
    const float* W1  = (const float*)d_in[20]; const float* b1  = (const float*)d_in[21];
    const float* W2  = (const float*)d_in[22]; const float* b2  = (const float*)d_in[23];
    const float* g2h = (const float*)d_in[24]; const float* be2h = (const float*)d_in[25];

    const int N = in_sizes[0] / DD;
    const int E = in_sizes[1] / DD;

    // ---- workspace layout (floats) ----
    float* ws = (float*)d_ws;
    size_t off = 0;
    auto alloc = [&](size_t n) { float* p = ws + off; off += n; return p; };
    float* deg   = alloc(N);
    float* Q     = alloc((size_t)N * DD);
    float* Kf    = alloc((size_t)N * DD);
    float* V     = alloc((size_t)N * DD);
    float* Ee    = alloc((size_t)E * 128);
    float* wE    = alloc((size_t)E * DD);
    float* m     = alloc((size_t)N * HH);
    float* denom = alloc((size_t)N * HH);
    float* wV    = alloc((size_t)N * DD);
    float* rowV  = alloc((size_t)N * DD);
    float* hmid  = alloc((size_t)N * DD);
    float* hO    = alloc((size_t)N * DD);
    float* hbn1  = alloc((size_t)N * DD);
    float* t1    = alloc((size_t)N * 128);
    float* h2    = alloc((size_t)N * DD);
    float* eOe   = alloc((size_t)E * DD);
    float* stats = alloc(256);

    float* h_out = (float*)d_out;
    float* e_out = (float*)d_out + (size_t)N * DD;

    auto strip_blocks = [](int M) { return (((M + 15) / 16) + 7) / 8; };  // 8 waves/block

    // ---- init accumulators ----
    fill_kernel<<<eb(N), 256, 0, stream>>>(deg, 0.f, N);
    fill_kernel<<<eb(N * HH), 256, 0, stream>>>(m, -INFINITY, N * HH);
    fill_kernel<<<eb(N * HH), 256, 0, stream>>>(denom, 0.f, N * HH);
    fill_kernel<<<eb(N * DD), 256, 0, stream>>>(wV, 0.f, N * DD);
    fill_kernel<<<eb(N * DD), 256, 0, stream>>>(rowV, 0.f, N * DD);

    // ---- degrees ----
    deg_kernel<<<eb(E), 256, 0, stream>>>(ei, deg, E);

    // ---- projections (WMMA fp32) ----
    gemm_wmma<64, 64, true,  false><<<strip_blocks(N), 256, 0, stream>>>(x, Wq, bq, Q, N);
    gemm_wmma<64, 64, false, false><<<strip_blocks(N), 256, 0, stream>>>(x, Wk, nullptr, Kf, N);
    gemm_wmma<64, 64, false, false><<<strip_blocks(N), 256, 0, stream>>>(x, Wv, nullptr, V, N);
    gemm_wmma<64, 128, true, false><<<strip_blocks(E), 256, 0, stream>>>(edge_attr, We, be, Ee, E);

    // ---- edge attention ----
    edge_s_kernel<<<eb(E * DD), 256, 0, stream>>>(ei, Q, Kf, Ee, wE, E);
    score_max_kernel<<<eb(E * HH), 256, 0, stream>>>(ei, wE, Aw, m, E);
    m_fix_kernel<<<eb(N * HH), 256, 0, stream>>>(m, N * HH);
    denom_kernel<<<eb(E * HH), 256, 0, stream>>>(ei, wE, Aw, m, denom, E);
    aggregate_kernel<<<eb(E * HH), 256, 0, stream>>>(ei, wE, Aw, m, denom, V, wV, rowV, E);
    node_kernel<<<eb(N * DD), 256, 0, stream>>>(wV, rowV, VeRow, deg, deg_coef, hmid, N);

    // ---- node output projection + BN1h ----
    gemm_wmma<64, 64, true, false><<<strip_blocks(N), 256, 0, stream>>>(hmid, WOh, bOh, hO, N);
    fill_kernel<<<1, 256, 0, stream>>>(stats, 0.f, 256);
    bn_stats_kernel<<<2048, 256, 0, stream>>>(x, hO, stats, N * DD);
    bn_finalize_kernel<<<1, 64, 0, stream>>>(stats, 1.0f / (float)N);
    bn_apply_kernel<<<eb(N * DD), 256, 0, stream>>>(x, hO, stats, g1h, be1h, hbn1, N * DD);

    // ---- edge output projection + BN1e -> e_out ----
    gemm_wmma<64, 64, true, false><<<strip_blocks(E), 256, 0, stream>>>(wE, WOe, bOe, eOe, E);
    fill_kernel<<<1, 256, 0, stream>>>(stats, 0.f, 256);
    bn_stats_kernel<<<4096, 256, 0, stream>>>(edge_attr, eOe, stats, E * DD);
    bn_finalize_kernel<<<1, 64, 0, stream>>>(stats, 1.0f / (float)E);
    bn_apply_kernel<<<eb(E * DD), 256, 0, stream>>>(edge_attr, eOe, stats, g1e, be1e, e_out, E * DD);

    // ---- FFN + BN2 -> h_out ----
    gemm_wmma<64, 128, true, true ><<<strip_blocks(N), 256, 0, stream>>>(hbn1, W1, b1, t1, N);
    gemm_wmma<128, 64, true, false><<<strip_blocks(N), 256, 0, stream>>>(t1, W2, b2, h2, N);
    fill_kernel<<<1, 256, 0, stream>>>(stats, 0.f, 256);
    bn_stats_kernel<<<2048, 256, 0, stream>>>(hbn1, h2, stats, N * DD);
    bn_finalize_kernel<<<1, 64, 0, stream>>>(stats, 1.0f / (float)N);
    bn_apply_kernel<<<eb(N * DD), 256, 0, stream>>>(hbn1, h2, stats, g2h, be2h, h_out, N * DD);
}